// QuanvolutionNet_65481071404085
// MI455X (gfx1250) — compile-verified
//
#include <hip/hip_runtime.h>
#include <math.h>

// ---------------------------------------------------------------------------
// Fused QuanvolutionNet forward for MI455X (gfx1250, wave32).
//   Phase A : 4-qubit statevector circuits in registers -> feats tile in LDS
//   Phase B : fp32 WMMA (v_wmma_f32_16x16x4_f32) projection 16x784 @ 784x16
//             (K-loop unrolled x2, dual accumulators for latency hiding)
//   Phase C : cross-wave reduce + bias + log_softmax -> out
// One kernel, zero HBM traffic for the 6.4MB intermediate feature tensor.
// ---------------------------------------------------------------------------

typedef __attribute__((ext_vector_type(2))) float v2f;
typedef __attribute__((ext_vector_type(8))) float v8f;

#define TNQ     4
#define TNP     196            // 14*14 patches per image
#define KDIM    784            // NP*NQ
#define KSTEPS  (KDIM / 4)     // 196 WMMA k-steps
#define MROWS   16             // images per workgroup (M tile)
#define THREADS 256            // 8 waves of 32
#define FPAD    788            // feats LDS row pitch (floats): 788%64=20 -> conflict-free
#define WPITCH  17             // Wt LDS row pitch: 17 -> conflict-free scatter/gather

// LDS carve-up (units: floats). Partials/cmat overlay earlier regions after barriers.
#define SM_FEATS 0
#define SM_WT    (MROWS * FPAD)               // 12608
#define SM_TOTAL (SM_WT + KDIM * WPITCH)      // 25936 floats = 103,744 B
#define SM_PART  SM_FEATS                     // 8*256 = 2048 floats (overlay feats)
#define SM_CM    SM_WT                        // 16*16 floats (overlay Wt)

__device__ __forceinline__ void apply_ry(float st[16], float c, float s, int mask) {
#pragma unroll
  for (int i = 0; i < 16; ++i) {
    if (i & mask) continue;
    const int j = i | mask;
    const float a0 = st[i], a1 = st[j];
    st[i] = c * a0 - s * a1;
    st[j] = s * a0 + c * a1;
  }
}

__device__ __forceinline__ void apply_cnot(float st[16], int cm, int tm) {
#pragma unroll
  for (int i = 0; i < 16; ++i) {
    if ((i & cm) && !(i & tm)) {
      const int j = i | tm;
      const float t = st[i]; st[i] = st[j]; st[j] = t;
    }
  }
}

// One 16x16x4 fp32 WMMA k-step: A from feats tile, B from W^T tile.
__device__ __forceinline__ v8f wmma_step(const float* __restrict__ smem,
                                         int mrow, int khalf, int s4, v8f acc) {
  const int kb = s4 * 4;
  // A fragment: lane -> row mrow, K = kb + 2*khalf + {0,1}  (contiguous b64)
  const float* fa = smem + SM_FEATS + mrow * FPAD + kb + 2 * khalf;
  v2f a; a.x = fa[0]; a.y = fa[1];
  // B fragment: VGPR0 holds K=kb(+2), VGPR1 holds K=kb+1(+3), N = mrow
  const float* fb = smem + SM_WT + (kb + 2 * khalf) * WPITCH + mrow;
  v2f bf; bf.x = fb[0]; bf.y = fb[WPITCH];
  return __builtin_amdgcn_wmma_f32_16x16x4_f32(
      /*neg_a=*/false, a, /*neg_b=*/false, bf,
      /*c_mod=*/(short)0, acc, /*reuse_a=*/false, /*reuse_b=*/false);
}

__global__ __launch_bounds__(THREADS)
void quanv_fused_kernel(const float* __restrict__ x,
                        const float* __restrict__ params,
                        const float* __restrict__ W,
                        const float* __restrict__ bias,
                        float* __restrict__ out) {
  __shared__ float smem[SM_TOTAL];
  const int tid  = threadIdx.x;
  const int img0 = blockIdx.x * MROWS;

  // Per-thread trig of the 8 circuit parameters (wave-uniform, tiny).
  float pc[2][4], ps[2][4];
#pragma unroll
  for (int l = 0; l < 2; ++l)
#pragma unroll
    for (int q = 0; q < 4; ++q) {
      const float h = 0.5f * params[l * 4 + q];
      pc[l][q] = __cosf(h);
      ps[l][q] = __sinf(h);
    }

  // ---------------- Phase A: circuits -> feats tile in LDS -----------------
  for (int idx = tid; idx < MROWS * TNP; idx += THREADS) {
    const int im = idx / TNP;
    const int pa = idx - im * TNP;       // patch id: p*14 + q
    const int p  = pa / 14;
    const int q  = pa - p * 14;
    const float* px = x + (size_t)(img0 + im) * 784 + (2 * p) * 28 + 2 * q;
    const float2 r0 = *(const float2*)(px);        // angles 0,1 (row 2p)
    const float2 r1 = *(const float2*)(px + 28);   // angles 2,3 (row 2p+1)

    float c[4], s[4];
    {
      const float ang[4] = { r0.x, r0.y, r1.x, r1.y };
#pragma unroll
      for (int w = 0; w < 4; ++w) {
        const float h = 0.5f * ang[w];
        c[w] = __cosf(h);
        s[w] = __sinf(h);
      }
    }

    // Input RYs on |0000> give a product state. Index = q0*8+q1*4+q2*2+q3.
    float st[16];
    {
      const float t01[4] = { c[0]*c[1], c[0]*s[1], s[0]*c[1], s[0]*s[1] };
      const float t23[4] = { c[2]*c[3], c[2]*s[3], s[2]*c[3], s[2]*s[3] };
#pragma unroll
      for (int i = 0; i < 16; ++i) st[i] = t01[i >> 2] * t23[i & 3];
    }

    // Two variational layers: RY(params) on each wire, then CNOT ring.
#pragma unroll
    for (int l = 0; l < 2; ++l) {
#pragma unroll
      for (int w = 0; w < 4; ++w) apply_ry(st, pc[l][w], ps[l][w], 8 >> w);
      apply_cnot(st, 8, 4);   // CNOT(0,1)
      apply_cnot(st, 4, 2);   // CNOT(1,2)
      apply_cnot(st, 2, 1);   // CNOT(2,3)
      apply_cnot(st, 1, 8);   // CNOT(3,0)
    }

    // <Z_w> = sum probs * (+1/-1 on bit of wire w)
    float z[4] = {0.f, 0.f, 0.f, 0.f};
#pragma unroll
    for (int i = 0; i < 16; ++i) {
      const float pr = st[i] * st[i];
#pragma unroll
      for (int w = 0; w < 4; ++w) z[w] += (i & (8 >> w)) ? -pr : pr;
    }

    float* f = smem + SM_FEATS + im * FPAD + pa * 4;
    f[0] = z[0]; f[1] = z[1]; f[2] = z[2]; f[3] = z[3];
  }

  // ------------- Phase A2: W^T (zero-padded to 16 cols) -> LDS -------------
  for (int i = tid; i < KDIM * 16; i += THREADS) {
    const int n = i / KDIM;              // output class (0..15)
    const int k = i - n * KDIM;          // coalesced along k
    smem[SM_WT + k * WPITCH + n] = (n < 10) ? W[n * KDIM + k] : 0.0f;
  }
  __syncthreads();

  // ---------------- Phase B: fp32 WMMA over K, 8 waves ---------------------
  // Round-robin k-steps per wave; unrolled x2 with independent accumulators
  // so the 4 DS loads of a pair issue together and overlap the first WMMA.
  const int wave  = tid >> 5;
  const int lane  = tid & 31;
  const int mrow  = lane & 15;           // M (for A) / N (for B)
  const int khalf = lane >> 4;           // 0: K+{0,1}, 1: K+{2,3}
  v8f acc0 = {};
  v8f acc1 = {};
  int s4 = wave;
  for (; s4 + 8 < KSTEPS; s4 += 16) {
    acc0 = wmma_step(smem, mrow, khalf, s4,     acc0);
    acc1 = wmma_step(smem, mrow, khalf, s4 + 8, acc1);
  }
  if (s4 < KSTEPS)                       // waves 0-3 have 25 (odd) steps
    acc0 = wmma_step(smem, mrow, khalf, s4, acc0);
  v8f acc;
#pragma unroll
  for (int r = 0; r < 8; ++r) acc[r] = acc0[r] + acc1[r];
  __syncthreads();                       // all waves done reading feats/Wt

  // ---------------- Phase C: cross-wave reduce + log_softmax ---------------
  float* part = smem + SM_PART;
#pragma unroll
  for (int r = 0; r < 8; ++r) part[wave * 256 + lane * 8 + r] = acc[r];
  __syncthreads();

  {
    const int l2 = tid >> 3, r = tid & 7;
    float sum = 0.f;
#pragma unroll
    for (int w = 0; w < 8; ++w) sum += part[w * 256 + l2 * 8 + r];
    const int M = r + ((l2 >> 4) << 3);  // C/D layout: VGPR r, lanes 16-31 -> M+8
    const int N = l2 & 15;
    smem[SM_CM + M * 16 + N] = sum;
  }
  __syncthreads();

  if (tid < MROWS) {
    float lg[10];
    float mx = -1e30f;
#pragma unroll
    for (int n = 0; n < 10; ++n) {
      lg[n] = smem[SM_CM + tid * 16 + n] + bias[n];
      mx = fmaxf(mx, lg[n]);
    }
    float se = 0.f;
#pragma unroll
    for (int n = 0; n < 10; ++n) se += __expf(lg[n] - mx);
    const float ls = __logf(se) + mx;
    float* o = out + (size_t)(img0 + tid) * 10;
#pragma unroll
    for (int n = 0; n < 10; ++n) o[n] = lg[n] - ls;
  }
}

extern "C" void kernel_launch(void* const* d_in, const int* in_sizes, int n_in,
                              void* d_out, int out_size, void* d_ws, size_t ws_size,
                              hipStream_t stream) {
  (void)n_in; (void)d_ws; (void)ws_size; (void)out_size;
  const float* x      = (const float*)d_in[0];   // [B,1,28,28]
  const float* params = (const float*)d_in[1];   // [2,4]
  const float* W      = (const float*)d_in[2];   // [10,784]
  const float* bias   = (const float*)d_in[3];   // [10]
  float* out          = (float*)d_out;           // [B,10]

  const int Bn     = in_sizes[0] / 784;          // 2048
  const int nblock = Bn / MROWS;                 // 128
  quanv_fused_kernel<<<nblock, THREADS, 0, stream>>>(x, params, W, bias, out);
}